// TreeLoss_35862976921799
// MI455X (gfx1250) — compile-verified
//
#include <hip/hip_runtime.h>

// Tree softmax cross-entropy loss, MI455X (gfx1250, wave32).
// Pass 1: one wave32 per sample; G=64 group -> 2 slots/lane; shuffle reductions.
// Pass 2: deterministic fixed-order reduction of per-block partials.

#define WAVES_PER_BLOCK 8
#define BLOCK_THREADS   (WAVES_PER_BLOCK * 32)

__global__ __launch_bounds__(BLOCK_THREADS) void tree_loss_pass1(
    const float* __restrict__ cls_score,      // (N, V)
    const float* __restrict__ weight,         // (V,)
    const int*   __restrict__ label,          // (N,)
    const int*   __restrict__ level_idx,      // (V, L, G)
    const unsigned char* __restrict__ level_mask,  // (V, L, G) bool
    const int*   __restrict__ local_label,    // (V, L)
    const unsigned char* __restrict__ level_valid, // (V, L) bool
    float* __restrict__ partial,              // (gridDim.x,)
    int N, int V, int L, int G)
{
    __shared__ float wsum[WAVES_PER_BLOCK];

    const int lane = threadIdx.x & 31;
    const int wave = threadIdx.x >> 5;
    const int i    = blockIdx.x * WAVES_PER_BLOCK + wave;

    float per_sample = 0.0f;

    if (i < N) {
        const int lbl = label[i];                      // uniform across wave
        const float* __restrict__ row = cls_score + (size_t)i * (size_t)V;

        float ce_sum = 0.0f;
        int   k      = 0;

        for (int l = 0; l < L; ++l) {
            const int vl = lbl * L + l;
            if (!level_valid[vl]) break;               // valid levels are a prefix
            ++k;

            const int ll   = local_label[vl];
            const int base = vl * G;

            // two group slots per lane (G = 64, wave32)
            const int g0 = lane;
            const int g1 = lane + 32;

            // coalesced index/mask loads (tables are L2-resident: ~4 MB)
            const int  i0 = level_idx[base + g0];
            const bool m0 = level_mask[base + g0] != 0;
            int  i1 = 0;
            bool m1 = false;
            if (g1 < G) {
                i1 = level_idx[base + g1];
                m1 = level_mask[base + g1] != 0;
            }

            // EXEC-predicated gathers: only fetch unmasked logits (saves HBM)
            float v0 = -1e30f, v1 = -1e30f;
            if (m0) v0 = row[i0];
            if (m1) v1 = row[i1];

            // wave32 max-reduction
            float mx = fmaxf(v0, v1);
            #pragma unroll
            for (int off = 16; off > 0; off >>= 1)
                mx = fmaxf(mx, __shfl_xor(mx, off, 32));

            // joint sum-reduction: exp-sum and target pick
            float s = expf(v0 - mx) + expf(v1 - mx);   // masked slots underflow to 0
            float t = (g0 == ll ? v0 : 0.0f) + (g1 == ll ? v1 : 0.0f);
            #pragma unroll
            for (int off = 16; off > 0; off >>= 1) {
                s += __shfl_xor(s, off, 32);
                t += __shfl_xor(t, off, 32);
            }

            ce_sum += (mx + logf(s)) - t;              // lse - tgt
        }

        per_sample = (ce_sum / (float)k) * weight[lbl];
    }

    if (lane == 0) wsum[wave] = per_sample;
    __syncthreads();

    // deterministic fixed-order block reduction (8 values)
    if (threadIdx.x == 0) {
        float acc = 0.0f;
        #pragma unroll
        for (int w = 0; w < WAVES_PER_BLOCK; ++w) acc += wsum[w];
        partial[blockIdx.x] = acc;
    }
}

__global__ __launch_bounds__(256) void tree_loss_pass2(
    const float* __restrict__ partial, float* __restrict__ out,
    int n_partial, float scale)
{
    __shared__ float sm[256];
    float acc = 0.0f;
    for (int t = threadIdx.x; t < n_partial; t += 256)
        acc += partial[t];                 // fixed strided order -> deterministic
    sm[threadIdx.x] = acc;
    __syncthreads();
    #pragma unroll
    for (int off = 128; off > 0; off >>= 1) {
        if ((int)threadIdx.x < off) sm[threadIdx.x] += sm[threadIdx.x + off];
        __syncthreads();
    }
    if (threadIdx.x == 0) out[0] = sm[0] * scale;
}

extern "C" void kernel_launch(void* const* d_in, const int* in_sizes, int n_in,
                              void* d_out, int out_size, void* d_ws, size_t ws_size,
                              hipStream_t stream) {
    const float*         cls_score   = (const float*)d_in[0];
    const float*         weight      = (const float*)d_in[1];
    const int*           label       = (const int*)d_in[2];
    const int*           level_idx   = (const int*)d_in[3];
    const unsigned char* level_mask  = (const unsigned char*)d_in[4];
    const int*           local_label = (const int*)d_in[5];
    const unsigned char* level_valid = (const unsigned char*)d_in[6];

    const int N = in_sizes[2];                 // label: (N,)
    const int V = in_sizes[1];                 // weight: (V,)
    const int L = in_sizes[5] / V;             // local_label: (V, L)
    const int G = in_sizes[3] / (V * L);       // level_idx: (V, L, G)

    const int blocks = (N + WAVES_PER_BLOCK - 1) / WAVES_PER_BLOCK;
    float* partial = (float*)d_ws;             // blocks * 4 bytes (16 KB @ N=32768)

    tree_loss_pass1<<<blocks, BLOCK_THREADS, 0, stream>>>(
        cls_score, weight, label, level_idx, level_mask, local_label, level_valid,
        partial, N, V, L, G);

    const float scale = 1.0f / (float)N;       // LOSS_WEIGHT == 1.0
    tree_loss_pass2<<<1, 256, 0, stream>>>(partial, (float*)d_out, blocks, scale);
}